// QGuidedDrifting_87703232184995
// MI455X (gfx1250) — compile-verified
//
#include <hip/hip_runtime.h>

// Problem constants (from reference)
#define BB 1024
#define QQ 128
#define PP 32
#define NN 128
#define DD 64
#define KT 160           // NN + PP (concat order: [neg | pos])
#define XS 68            // LDS row stride (floats) for x/y tiles (bank-conflict pad)
#define WSTR 164         // LDS row stride (floats) for dist/affinity matrices

#define LOG2E 1.4426950408889634f

typedef __attribute__((ext_vector_type(2))) float v2f;
typedef __attribute__((ext_vector_type(8))) float v8f;

// ---------------------------------------------------------------------------
// Phase 1: per-batch partial sum of ||x_q - ypos_p|| over (q,p)
// ---------------------------------------------------------------------------
__global__ void __launch_bounds__(256) k_meandist(const float* __restrict__ x,
                                                  const float* __restrict__ yp,
                                                  float* __restrict__ partial) {
  __shared__ float sYp[PP * XS];
  __shared__ float red[256];
  const int b = blockIdx.x, tid = threadIdx.x;
  const float* xb  = x  + (size_t)b * QQ * DD;
  const float* ypb = yp + (size_t)b * PP * DD;

  for (int idx = tid; idx < PP * DD / 4; idx += 256) {
    const int p = idx >> 4, dv = idx & 15;
    *(float4*)&sYp[p * XS + dv * 4] = ((const float4*)ypb)[idx];
  }
  __syncthreads();

  float acc = 0.f;
  const int lane = tid & 31;
  for (int i = 0; i < 16; ++i) {
    const int q = (i << 3) + (tid >> 5);   // one wave handles one q, 32 p's
    const float* xr = xb + q * DD;
    const float* yr = &sYp[lane * XS];
    float d2 = 0.f;
    for (int d = 0; d < DD; d += 4) {
      const float4 xv = *(const float4*)&xr[d];
      const float e0 = xv.x - yr[d + 0];
      const float e1 = xv.y - yr[d + 1];
      const float e2 = xv.z - yr[d + 2];
      const float e3 = xv.w - yr[d + 3];
      d2 += e0 * e0 + e1 * e1 + e2 * e2 + e3 * e3;
    }
    acc += __builtin_amdgcn_sqrtf(d2);
  }
  red[tid] = acc;
  __syncthreads();
  for (int s = 128; s > 0; s >>= 1) {
    if (tid < s) red[tid] += red[tid + s];
    __syncthreads();
  }
  if (tid == 0) partial[b] = red[0];
}

// Deterministic fixed-order reduction of n partials -> out[0] = sum * scale
__global__ void __launch_bounds__(256) k_reduce(const float* __restrict__ partial,
                                                int n, float scale,
                                                float* __restrict__ out) {
  __shared__ float red[256];
  const int tid = threadIdx.x;
  float a = 0.f;
  for (int i = tid; i < n; i += 256) a += partial[i];
  red[tid] = a;
  __syncthreads();
  for (int s = 128; s > 0; s >>= 1) {
    if (tid < s) red[tid] += red[tid + s];
    __syncthreads();
  }
  if (tid == 0) out[0] = red[0] * scale;
}

// ---------------------------------------------------------------------------
// Phase 2: per-batch drift over 3 temperatures + partial loss
// ---------------------------------------------------------------------------
static constexpr int SMEM_FLOATS =
    QQ * XS + NN * XS + PP * XS   // sX, sYn, sYp
    + 2 * QQ * WSTR               // sDist, sW (affinity)
    + QQ + KT                     // sX2, sY2
    + QQ + KT                     // rs, cs
    + QQ + QQ                     // rsn, rsp
    + 256;                        // red

__global__ void __launch_bounds__(256, 1) k_drift(const float* __restrict__ x,
                                                  const float* __restrict__ yp,
                                                  const float* __restrict__ yn,
                                                  const float* __restrict__ meanDist,
                                                  float* __restrict__ partialLoss) {
  extern __shared__ float smem[];
  float* sX    = smem;               // QQ x XS
  float* sYn   = sX + QQ * XS;       // NN x XS
  float* sYp   = sYn + NN * XS;      // PP x XS
  float* sDist = sYp + PP * XS;      // QQ x WSTR
  float* sW    = sDist + QQ * WSTR;  // QQ x WSTR (plain affinity)
  float* sX2   = sW + QQ * WSTR;     // QQ
  float* sY2   = sX2 + QQ;           // KT  ([neg | pos])
  float* rs    = sY2 + KT;           // QQ
  float* cs    = rs + QQ;            // KT
  float* rsn   = cs + KT;            // QQ
  float* rsp   = rsn + QQ;           // QQ
  float* red   = rsp + QQ;           // 256

  const int b = blockIdx.x, tid = threadIdx.x;
  const int wv = tid >> 5, lane = tid & 31, l16 = lane & 15, hi = lane >> 4;
  const int m0 = wv * 16;  // this wave's 16-row strip

  const float* xb  = x  + (size_t)b * QQ * DD;
  const float* ypb = yp + (size_t)b * PP * DD;
  const float* ynb = yn + (size_t)b * NN * DD;

  // ---- stage inputs in LDS (float4, coalesced) ----
  for (int idx = tid; idx < QQ * DD / 4; idx += 256) {
    const int q = idx >> 4, dv = idx & 15;
    *(float4*)&sX[q * XS + dv * 4]  = ((const float4*)xb)[idx];
    *(float4*)&sYn[q * XS + dv * 4] = ((const float4*)ynb)[idx];
  }
  for (int idx = tid; idx < PP * DD / 4; idx += 256) {
    const int p = idx >> 4, dv = idx & 15;
    *(float4*)&sYp[p * XS + dv * 4] = ((const float4*)ypb)[idx];
  }
  __syncthreads();

  // ---- row norms ----
  if (tid < QQ) {
    const float* r = &sX[tid * XS];
    float s = 0.f;
    for (int d = 0; d < DD; ++d) s += r[d] * r[d];
    sX2[tid] = s;
  }
  if (tid < KT) {
    const float* r = (tid < NN) ? &sYn[tid * XS] : &sYp[(tid - NN) * XS];
    float s = 0.f;
    for (int d = 0; d < DD; ++d) s += r[d] * r[d];
    sY2[tid] = s;
  }
  __syncthreads();

  // ---- Gram via FP32 WMMA: k outer so the A fragment is loaded once and
  //      shared by all 10 column tiles (10 independent WMMA chains) ----
  {
    v8f cg[10];
#pragma unroll
    for (int t = 0; t < 10; ++t)
#pragma unroll
      for (int r = 0; r < 8; ++r) cg[t][r] = 0.f;

    const float* xrow = &sX[(m0 + l16) * XS];
#pragma unroll
    for (int k0 = 0; k0 < DD; k0 += 4) {
      const v2f a = *(const v2f*)&xrow[k0 + 2 * hi];    // A[m][k]
#pragma unroll
      for (int t = 0; t < 10; ++t) {
        const float* ybase = (t < 8) ? &sYn[(t * 16 + l16) * XS]
                                     : &sYp[((t - 8) * 16 + l16) * XS];
        const v2f bb = *(const v2f*)&ybase[k0 + 2 * hi]; // B[k][n] = Y[n][k]
        cg[t] = __builtin_amdgcn_wmma_f32_16x16x4_f32(false, a, false, bb,
                                                      (short)0, cg[t], false, false);
      }
    }
    // dist = sqrt(max(x2 + y2 - 2*S, 0))
#pragma unroll
    for (int t = 0; t < 10; ++t) {
      const int col = t * 16 + l16;
      const float y2 = sY2[col];
#pragma unroll
      for (int r = 0; r < 8; ++r) {
        const int row = m0 + r + 8 * hi;
        const float d2 = sX2[row] + y2 - 2.f * cg[t][r];
        sDist[row * WSTR + col] = __builtin_amdgcn_sqrtf(fmaxf(d2, 0.f));
      }
    }
  }

  const float mD = meanDist[0];
  const float temps[3] = {0.1f, 0.5f, 1.0f};

  v8f acc[4];  // drift accumulators: D cols 0..63 in 4 tiles of 16
#pragma unroll
  for (int t = 0; t < 4; ++t)
#pragma unroll
    for (int r = 0; r < 8; ++r) acc[t][r] = 0.f;

  for (int ti = 0; ti < 3; ++ti) {
    const float at = fmaxf(temps[ti] * mD, 1e-6f);
    const float c2 = LOG2E / at;           // exp(max(-d/t,-80)) = exp2(max(-d*c2, -80*log2e))
    const float clampv = -80.f * LOG2E;

    __syncthreads();  // dist complete (iter 0) / prev WMMA done before sW rewrite

    // pass A: row sums and column sums of the exp kernel
    if (tid < QQ) {
      const float* dr = &sDist[tid * WSTR];
      float s = 0.f;
      for (int k = 0; k < KT; ++k)
        s += __builtin_amdgcn_exp2f(fmaxf(-dr[k] * c2, clampv));
      rs[tid] = s;
    }
    if (tid < KT) {
      float s = 0.f;
      for (int q = 0; q < QQ; ++q)
        s += __builtin_amdgcn_exp2f(fmaxf(-sDist[q * WSTR + tid] * c2, clampv));
      cs[tid] = s;
    }
    __syncthreads();

    // pass B: affinity into sW + pos/neg row sums (row scaling deferred to
    // the A-fragment multiply in the WMMA loop below)
    if (tid < QQ) {
      float* wr = &sW[tid * WSTR];
      const float* dr = &sDist[tid * WSTR];
      const float rq = rs[tid];
      float sn = 0.f, sp = 0.f;
      for (int k = 0; k < KT; ++k) {
        const float kv = __builtin_amdgcn_exp2f(fmaxf(-dr[k] * c2, clampv));
        const float aff = kv * __builtin_amdgcn_rsqf(fmaxf(rq * cs[k], 1e-12f));
        wr[k] = aff;
        if (k < NN) sn += aff; else sp += aff;
      }
      rsn[tid] = sn;
      rsp[tid] = sp;
    }
    __syncthreads();

    // drift += [-aff_neg*rsp | aff_pos*rsn] @ [y_neg; y_pos]  via FP32 WMMA.
    // k outer: A fragment loaded once, scaled in registers, shared by 4 tiles.
    const float* arow = &sW[(m0 + l16) * WSTR];
    const float sNeg = -rsp[m0 + l16];   // per-lane row constant (A rows = m0+l16)
    const float sPos =  rsn[m0 + l16];

#pragma unroll 4
    for (int k0 = 0; k0 < NN; k0 += 4) {           // negative segment
      v2f a = *(const v2f*)&arow[k0 + 2 * hi];
      a *= sNeg;
      const float* r0 = &sYn[(k0 + 2 * hi) * XS];
      const float* r1 = r0 + XS;
#pragma unroll
      for (int t = 0; t < 4; ++t) {
        v2f bb;
        bb.x = r0[t * 16 + l16];
        bb.y = r1[t * 16 + l16];
        acc[t] = __builtin_amdgcn_wmma_f32_16x16x4_f32(false, a, false, bb,
                                                       (short)0, acc[t], false, false);
      }
    }
#pragma unroll 4
    for (int k0 = 0; k0 < PP; k0 += 4) {           // positive segment
      v2f a = *(const v2f*)&arow[NN + k0 + 2 * hi];
      a *= sPos;
      const float* r0 = &sYp[(k0 + 2 * hi) * XS];
      const float* r1 = r0 + XS;
#pragma unroll
      for (int t = 0; t < 4; ++t) {
        v2f bb;
        bb.x = r0[t * 16 + l16];
        bb.y = r1[t * 16 + l16];
        acc[t] = __builtin_amdgcn_wmma_f32_16x16x4_f32(false, a, false, bb,
                                                       (short)0, acc[t], false, false);
      }
    }
  }

  // ---- loss partial: sum of drift^2 over this block's batch ----
  float ls = 0.f;
#pragma unroll
  for (int t = 0; t < 4; ++t)
#pragma unroll
    for (int r = 0; r < 8; ++r) ls += acc[t][r] * acc[t][r];
  red[tid] = ls;
  __syncthreads();
  for (int s = 128; s > 0; s >>= 1) {
    if (tid < s) red[tid] += red[tid + s];
    __syncthreads();
  }
  if (tid == 0) partialLoss[b] = red[0];
}

// ---------------------------------------------------------------------------
extern "C" void kernel_launch(void* const* d_in, const int* in_sizes, int n_in,
                              void* d_out, int out_size, void* d_ws, size_t ws_size,
                              hipStream_t stream) {
  (void)in_sizes; (void)n_in; (void)out_size; (void)ws_size;
  const float* x  = (const float*)d_in[0];
  const float* yp = (const float*)d_in[1];
  const float* yn = (const float*)d_in[2];
  float* out = (float*)d_out;

  float* ws    = (float*)d_ws;
  float* partD = ws;             // [1024] per-batch dist sums
  float* meanD = ws + BB;        // [1]    mean distance
  float* partL = ws + BB + 1;    // [1024] per-batch loss sums

  k_meandist<<<BB, 256, 0, stream>>>(x, yp, partD);
  k_reduce<<<1, 256, 0, stream>>>(partD, BB,
                                  1.0f / ((float)BB * QQ * PP), meanD);

  const size_t shmem = (size_t)SMEM_FLOATS * sizeof(float);  // ~250 KB of 320 KB WGP LDS
  k_drift<<<BB, 256, shmem, stream>>>(x, yp, yn, meanD, partL);

  k_reduce<<<1, 256, 0, stream>>>(partL, BB,
                                  1.0f / ((float)BB * QQ * DD), out);
}